// PointRendROIHeads_18150531792987
// MI455X (gfx1250) — compile-verified
//
#include <hip/hip_runtime.h>
#include <hip/hip_bf16.h>

typedef __attribute__((ext_vector_type(2))) float v2f;
typedef __attribute__((ext_vector_type(8))) float v8f;

#define RR   64
#define KCLS 81
#define CFD  256
#define CIN  337
#define KPAD 340
#define NP   784
#define PTOT (RR * NP)   // 50176

// ---------------------------------------------------------------------------
// Step-1: all 28x28 points are selected; emit idx + normalized coords.
// ---------------------------------------------------------------------------
__global__ void k_init_points28(int* idx, float* coords) {
    int p = blockIdx.x * blockDim.x + threadIdx.x;
    if (p >= PTOT) return;
    int j = p % NP;
    idx[p] = j;
    coords[2 * p + 0] = ((j % 28) + 0.5f) / 28.0f;
    coords[2 * p + 1] = ((j / 28) + 0.5f) / 28.0f;
}

// ---------------------------------------------------------------------------
// 2x bilinear upsample, jax.image.resize('linear') semantics:
// src coord = out/2 - 0.25, edge clamped.
// ---------------------------------------------------------------------------
__global__ void k_upsample2x(const float* __restrict__ in, float* __restrict__ out,
                             int Hi, int Wi) {
    int Ho = 2 * Hi, Wo = 2 * Wi;
    size_t t = (size_t)blockIdx.x * blockDim.x + threadIdx.x;
    size_t total = (size_t)RR * KCLS * Ho * Wo;
    if (t >= total) return;
    int w = (int)(t % Wo);
    int h = (int)((t / Wo) % Ho);
    size_t c = t / ((size_t)Wo * Ho);
    const float* ip = in + c * (size_t)(Hi * Wi);
    float y = 0.5f * h - 0.25f;
    float x = 0.5f * w - 0.25f;
    float y0f = floorf(y), x0f = floorf(x);
    float fy = y - y0f, fx = x - x0f;
    int y0 = (int)y0f, x0 = (int)x0f;
    int y0c = min(max(y0, 0), Hi - 1);
    int y1c = min(max(y0 + 1, 0), Hi - 1);
    int x0c = min(max(x0, 0), Wi - 1);
    int x1c = min(max(x0 + 1, 0), Wi - 1);
    float v00 = ip[y0c * Wi + x0c], v01 = ip[y0c * Wi + x1c];
    float v10 = ip[y1c * Wi + x0c], v11 = ip[y1c * Wi + x1c];
    out[t] = (1.0f - fy) * ((1.0f - fx) * v00 + fx * v01) +
             fy * ((1.0f - fx) * v10 + fx * v11);
}

// ---------------------------------------------------------------------------
// Exact top-NP selection per ROI: sort (|gt| bits, idx) ascending in LDS
// (bitonic). Matches jax.lax.top_k tie behavior (lowest index wins).
// One 1024-thread workgroup per ROI; dynamic LDS = NPADN * 8 bytes
// (128KB for 16384 — only possible with CDNA5's 320KB WGP LDS).
// ---------------------------------------------------------------------------
template <int NPADN>
__global__ void k_select(const float* __restrict__ ml, const int* __restrict__ labels,
                         int Hd, int Wd, int* __restrict__ idx, float* __restrict__ coords) {
    extern __shared__ unsigned long long s[];
    int r = blockIdx.x;
    int HW = Hd * Wd;
    const float* base = ml + ((size_t)r * KCLS + (size_t)(labels[r] + 1)) * (size_t)HW;
    for (int i = threadIdx.x; i < NPADN; i += blockDim.x) {
        unsigned long long v;
        if (i < HW) {
            unsigned key = __float_as_uint(fabsf(base[i]));
            v = ((unsigned long long)key << 32) | (unsigned)i;
        } else {
            v = ~0ull;
        }
        s[i] = v;
    }
    __syncthreads();
    for (int k = 2; k <= NPADN; k <<= 1) {
        for (int j = k >> 1; j > 0; j >>= 1) {
            for (int i = threadIdx.x; i < NPADN; i += blockDim.x) {
                int ixj = i ^ j;
                if (ixj > i) {
                    unsigned long long a = s[i], b = s[ixj];
                    bool up = ((i & k) == 0);
                    if ((a > b) == up) { s[i] = b; s[ixj] = a; }
                }
            }
            __syncthreads();
        }
    }
    if (threadIdx.x < NP) {
        unsigned iv = (unsigned)s[threadIdx.x];
        int p = r * NP + threadIdx.x;
        idx[p] = (int)iv;
        coords[2 * p + 0] = ((iv % (unsigned)Wd) + 0.5f) / (float)Wd;
        coords[2 * p + 1] = ((iv / (unsigned)Wd) + 0.5f) / (float)Hd;
    }
}

// ---------------------------------------------------------------------------
// Fine feature sampling: grid_sample(feat[1,256,256,256], bilinear,
// zeros padding, align_corners=False). Block = point, thread = channel.
// feat stays L2-resident (67MB < 192MB L2).
// ---------------------------------------------------------------------------
__global__ void k_fine(const float* __restrict__ feat, const float* __restrict__ bboxes,
                       const float* __restrict__ coords, float* __restrict__ F) {
    int p = blockIdx.x;
    int c = threadIdx.x;
    int r = p / NP;
    float px = coords[2 * p], py = coords[2 * p + 1];
    float bx0 = bboxes[4 * r + 0], by0 = bboxes[4 * r + 1];
    float bw = bboxes[4 * r + 2] - bx0;
    float bh = bboxes[4 * r + 3] - by0;
    float cx = bx0 + px * bw;
    float cy = by0 + py * bh;
    // fcoords * W - 0.5 with W=H=256, fcoords = c*SCALE/256
    float x = cx * 0.25f - 0.5f;
    float y = cy * 0.25f - 0.5f;
    float x0f = floorf(x), y0f = floorf(y);
    float fx = x - x0f, fy = y - y0f;
    int x0 = (int)x0f, y0 = (int)y0f;
    const float* ch = feat + (size_t)c * (256 * 256);
    float acc = 0.0f;
#pragma unroll
    for (int dy = 0; dy < 2; ++dy) {
#pragma unroll
        for (int dx = 0; dx < 2; ++dx) {
            int ix = x0 + dx, iy = y0 + dy;
            float w = (dx ? fx : 1.0f - fx) * (dy ? fy : 1.0f - fy);
            bool valid = (ix >= 0) && (ix < 256) && (iy >= 0) && (iy < 256);
            int ixc = min(max(ix, 0), 255);
            int iyc = min(max(iy, 0), 255);
            acc += ch[iyc * 256 + ixc] * (valid ? w : 0.0f);
        }
    }
    F[(size_t)c * PTOT + p] = acc;
}

// ---------------------------------------------------------------------------
// Coarse sampling: grid_sample(coarse_logits[R,81,7,7]) at (px,py).
// ---------------------------------------------------------------------------
__global__ void k_coarse(const float* __restrict__ clog, const float* __restrict__ coords,
                         float* __restrict__ C) {
    int p = blockIdx.x;
    int c = threadIdx.x;
    if (c >= KCLS) return;
    int r = p / NP;
    float px = coords[2 * p], py = coords[2 * p + 1];
    float x = px * 7.0f - 0.5f;
    float y = py * 7.0f - 0.5f;
    float x0f = floorf(x), y0f = floorf(y);
    float fx = x - x0f, fy = y - y0f;
    int x0 = (int)x0f, y0 = (int)y0f;
    const float* ch = clog + ((size_t)r * KCLS + c) * 49;
    float acc = 0.0f;
#pragma unroll
    for (int dy = 0; dy < 2; ++dy) {
#pragma unroll
        for (int dx = 0; dx < 2; ++dx) {
            int ix = x0 + dx, iy = y0 + dy;
            float w = (dx ? fx : 1.0f - fx) * (dy ? fy : 1.0f - fy);
            bool valid = (ix >= 0) && (ix < 7) && (iy >= 0) && (iy < 7);
            int ixc = min(max(ix, 0), 6);
            int iyc = min(max(iy, 0), 6);
            acc += ch[iyc * 7 + ixc] * (valid ? w : 0.0f);
        }
    }
    C[(size_t)c * PTOT + p] = acc;
}

// ---------------------------------------------------------------------------
// WMMA GEMM: out[M x P] = W[M x 337] @ concat(Bh[256 x P]; Bc[81 x P]) + bias
// fp32 via V_WMMA_F32_16X16X4_F32 (full precision path).
// Block = 8 waves / 128 points; wave owns a 16-point N-tile, loops 16 M-tiles.
// Panels staged into dynamic LDS via the CDNA5 async global->LDS path
// (GLOBAL_LOAD_ASYNC_TO_LDS_B128/B32, tracked by ASYNCcnt; no VGPR bounce).
// mode 0: out = relu(.) into outF [m][p] (in-place with Bh is safe: B fully
//         staged before epilogue; columns disjoint per wave/block).
// mode 1: scatter (c = m < 81) into ml[r][c][idx[p]].
// ---------------------------------------------------------------------------
__global__ void k_gemm(const float* __restrict__ Wt, const float* __restrict__ bias,
                       const float* __restrict__ Bh, const float* __restrict__ Bc,
                       float* __restrict__ outF,
                       float* __restrict__ outMl, const int* __restrict__ idx, int HW,
                       int Mtiles, int Mrows, int mode) {
    extern __shared__ float lds[];
    float* Bsh = lds;                       // 8 * KPAD * 16 floats
    float* Ash = lds + 8 * KPAD * 16;       // 16 * KPAD floats
    int tid = threadIdx.x;
    int wave = tid >> 5;
    int lane = tid & 31;
    int p0 = blockIdx.x * 128 + wave * 16;
    float* Bw = Bsh + wave * (KPAD * 16);

    // --- Stage B panel for this wave: rows 0..255 from Bh, 256..336 from Bc.
    // Async global->LDS, 16B per lane per op. Addresses are 16B aligned:
    // p0 is a multiple of 16, ws regions 16B aligned.
    unsigned bw_base = (unsigned)(uintptr_t)(void*)Bw;  // LDS byte offset
    {
        int q = lane & 3;                                // 4 lanes per row
        for (int k = (lane >> 2); k < CIN; k += 8) {
            const float* src = (k < CFD)
                ? (Bh + (size_t)k * PTOT + p0 + q * 4)
                : (Bc + (size_t)(k - CFD) * PTOT + p0 + q * 4);
            unsigned dst = bw_base + (unsigned)((k * 16 + q * 4) * 4);
            asm volatile("global_load_async_to_lds_b128 %0, %1, off"
                         :: "v"(dst), "v"(src) : "memory");
        }
        // zero-pad rows 337..339 (NaN-safe padding; A pad is also zero)
        if (lane < 12) {
            int k = CIN + (lane >> 2);
            int qq = lane & 3;
            float4 z = make_float4(0.f, 0.f, 0.f, 0.f);
            *(float4*)(Bw + (k * 16 + qq * 4)) = z;
        }
    }
    asm volatile("s_wait_asynccnt 0x0" ::: "memory");
    __syncthreads();

    // WMMA f32 16x16x4 fragment lane mapping (ISA 7.12.2):
    // A: lanes 0-15 row M=lane, regs (K=k0,k0+1); lanes 16-31 row M=lane-16, (K=k0+2,k0+3)
    // B: lanes 0-15 col N=lane, regs (K=k0,k0+1); lanes 16-31 col N=lane-16, (K=k0+2,k0+3)
    int rA = lane & 15;
    int koff = (lane >> 4) << 1;
    int nB = lane & 15;
    unsigned ash_base = (unsigned)(uintptr_t)(void*)Ash;

    for (int mt = 0; mt < Mtiles; ++mt) {
        int m0 = mt * 16;
        // Stage A panel (shared by all 8 waves), zero-pad K>=337 and M>=Mrows.
        // W rows are only 4B aligned (stride 337 floats) -> async B32.
        for (int e = tid; e < 16 * KPAD; e += 256) {
            int rr = e / KPAD;
            int k = e - rr * KPAD;
            int m = m0 + rr;
            if (m < Mrows && k < CIN) {
                const float* src = Wt + (size_t)m * CIN + k;
                unsigned dst = ash_base + (unsigned)(e * 4);
                asm volatile("global_load_async_to_lds_b32 %0, %1, off"
                             :: "v"(dst), "v"(src) : "memory");
            } else {
                Ash[e] = 0.0f;
            }
        }
        asm volatile("s_wait_asynccnt 0x0" ::: "memory");
        __syncthreads();

        v8f acc = {};
        const float* Arow = Ash + rA * KPAD + koff;
        const float* Brow = Bw + koff * 16 + nB;
        for (int kk = 0; kk < KPAD / 4; ++kk) {
            v2f a = *(const v2f*)(Arow + kk * 4);       // ds_read_b64
            v2f b;
            b.x = Brow[kk * 64];                        // ds_load_2addr_b32 pair
            b.y = Brow[kk * 64 + 16];
            acc = __builtin_amdgcn_wmma_f32_16x16x4_f32(
                false, a, false, b, (short)0, acc, false, false);
        }

        // Epilogue. C/D layout: VGPR j, lanes0-15 -> M=m0+j, lanes16-31 -> M=m0+j+8; N=lane&15.
        int mhalf = (lane >> 4) << 3;
        int n = lane & 15;
        int p = p0 + n;
        if (mode == 0) {
#pragma unroll
            for (int j = 0; j < 8; ++j) {
                int m = m0 + j + mhalf;
                float v = acc[j] + bias[m];
                outF[(size_t)m * PTOT + p] = fmaxf(v, 0.0f);
            }
        } else {
            int iv = idx[p];
            int r = p / NP;
#pragma unroll
            for (int j = 0; j < 8; ++j) {
                int m = m0 + j + mhalf;
                if (m < Mrows) {
                    float v = acc[j] + bias[m];
                    outMl[((size_t)r * KCLS + m) * (size_t)HW + iv] = v;
                }
            }
        }
        __syncthreads();
    }
}

// ---------------------------------------------------------------------------
// Host orchestration
// ---------------------------------------------------------------------------
extern "C" void kernel_launch(void* const* d_in, const int* in_sizes, int n_in,
                              void* d_out, int out_size, void* d_ws, size_t ws_size,
                              hipStream_t stream) {
    const float* feat   = (const float*)d_in[0];
    const float* bboxes = (const float*)d_in[1];
    const int*   labels = (const int*)d_in[2];
    const float* clog   = (const float*)d_in[3];
    const float* W1 = (const float*)d_in[4];
    const float* b1 = (const float*)d_in[5];
    const float* W2 = (const float*)d_in[6];
    const float* b2 = (const float*)d_in[7];
    const float* W3 = (const float*)d_in[8];
    const float* b3 = (const float*)d_in[9];
    const float* Wp = (const float*)d_in[10];
    const float* bp = (const float*)d_in[11];
    float* out = (float*)d_out;

    float* F      = (float*)d_ws;                       // 256 * PTOT (fine -> h1 -> h2 -> h3 in place)
    float* C      = F + (size_t)CFD * PTOT;             // 81 * PTOT
    int*   idx    = (int*)(C + (size_t)KCLS * PTOT);    // PTOT
    float* coords = (float*)(idx + PTOT);               // 2 * PTOT
    float* ml28   = coords + 2 * (size_t)PTOT;          // 64*81*784
    float* ml56   = ml28 + (size_t)RR * KCLS * 784;     // 64*81*3136

    size_t shm = (size_t)(8 * KPAD * 16 + 16 * KPAD) * sizeof(float);
    int nblk = PTOT / 128;  // 392

    auto run_head = [&](float* mlOut, int HW) {
        k_fine<<<PTOT, 256, 0, stream>>>(feat, bboxes, coords, F);
        k_coarse<<<PTOT, 96, 0, stream>>>(clog, coords, C);
        k_gemm<<<nblk, 256, shm, stream>>>(W1, b1, F, C, F, nullptr, nullptr, 0, 16, 256, 0);
        k_gemm<<<nblk, 256, shm, stream>>>(W2, b2, F, C, F, nullptr, nullptr, 0, 16, 256, 0);
        k_gemm<<<nblk, 256, shm, stream>>>(W3, b3, F, C, F, nullptr, nullptr, 0, 16, 256, 0);
        k_gemm<<<nblk, 256, shm, stream>>>(Wp, bp, F, C, nullptr, mlOut, idx, HW, 6, 81, 1);
    };

    // Step 1 (H=28): all points selected -> ml28 is fully the point-head output.
    k_init_points28<<<(PTOT + 255) / 256, 256, 0, stream>>>(idx, coords);
    run_head(ml28, 28 * 28);

    // Step 2 (H=56)
    {
        size_t tot = (size_t)RR * KCLS * 56 * 56;
        k_upsample2x<<<(int)((tot + 255) / 256), 256, 0, stream>>>(ml28, ml56, 28, 28);
        k_select<4096><<<RR, 1024, 4096 * sizeof(unsigned long long), stream>>>(
            ml56, labels, 56, 56, idx, coords);
        run_head(ml56, 56 * 56);
    }

    // Step 3 (H=112), upsample directly into d_out, refine in place.
    {
        size_t tot = (size_t)RR * KCLS * 112 * 112;
        k_upsample2x<<<(int)((tot + 255) / 256), 256, 0, stream>>>(ml56, out, 56, 56);
        k_select<16384><<<RR, 1024, 16384 * sizeof(unsigned long long), stream>>>(
            out, labels, 112, 112, idx, coords);
        run_head(out, 112 * 112);
    }
}